// GraphTopoAttention_86964497809756
// MI455X (gfx1250) — compile-verified
//
#include <hip/hip_runtime.h>

typedef __attribute__((ext_vector_type(2))) float v2f;
typedef __attribute__((ext_vector_type(8))) float v8f;

#define NN   50000
#define EE   800000
#define HH   4
#define INDIM 256
#define OC   384          // combined output cols: [ft(128) | ft_c(128) | ft_q(128)]
#define LSK  570400u      // int(0.713 * E)
#define OUTC 160

struct Ctl {
  unsigned prefix[4];   // words 0..3  : radix-select prefix -> final = key of thd
  unsigned krem[4];     // words 4..7  : remaining rank within prefix class
  unsigned cnt_lt[4];
  float    sum_lt[4];
  float    sum_tot[4];
  float    ratio[4];
  unsigned hist[4 * 256];
};

// monotone float->uint key (total order, handles negatives from ELU)
__device__ __forceinline__ unsigned fkey(float x) {
  unsigned u = __float_as_uint(x);
  return (u & 0x80000000u) ? ~u : (u | 0x80000000u);
}
__device__ __forceinline__ float fkey_inv(unsigned k) {
  return (k & 0x80000000u) ? __uint_as_float(k & 0x7fffffffu) : __uint_as_float(~k);
}

// ---------------- init: zero out, mkey, ssum, ctl (krem <- LSK) ----------------
__global__ void init_kernel(float* out, unsigned* mkey, float* ssum, Ctl* ctl) {
  int i = blockIdx.x * blockDim.x + threadIdx.x;
  int stride = gridDim.x * blockDim.x;
  for (int j = i; j < NN * OUTC; j += stride) out[j] = 0.0f;
  for (int j = i; j < NN * HH; j += stride) { mkey[j] = 0u; ssum[j] = 0.0f; }
  unsigned* cw = (unsigned*)ctl;
  int nw = (int)(sizeof(Ctl) / 4);
  for (int j = i; j < nw; j += stride) cw[j] = (j >= 4 && j < 8) ? LSK : 0u;
}

// ---------------- pack weights: unified B [256 x 384], K-pair interleaved ------
// Bp[(k>>1)*768 + col*2 + (k&1)] = B[k][col] so each lane's two K-values for a
// WMMA fragment are one contiguous b64.
__global__ void pack_kernel(const float* __restrict__ Wl, const float* __restrict__ fc,
                            const float* __restrict__ fq, float* __restrict__ Wp) {
  int gid = blockIdx.x * blockDim.x + threadIdx.x;
  if (gid >= INDIM * OC) return;
  int k = gid / OC, col = gid % OC;
  float v;
  if (col < 128)      v = Wl[col * INDIM + k];          // ft = h @ Wl.T
  else if (col < 256) v = fc[k * 128 + (col - 128)];    // ft_c = h @ fc
  else                v = fq[k * 128 + (col - 256)];    // ft_q = h @ fq
  Wp[(k >> 1) * (OC * 2) + col * 2 + (k & 1)] = v;
}

// ---------------- fused triple GEMM via V_WMMA_F32_16X16X4_F32 ----------------
// block = 256 threads (8 waves); block handles a 16-row tile of h; each wave
// computes three 16x16 output tiles (24 column tiles of y total).
// h-tile is staged into padded LDS with GLOBAL_LOAD_ASYNC_TO_LDS_B128.
__global__ __launch_bounds__(256) void gemm_kernel(const float* __restrict__ h,
                                                   const float* __restrict__ Wp,
                                                   float* __restrict__ y) {
  __shared__ float At[16 * 260];   // padded stride 260 -> conflict-free ds reads
  int rowbase = blockIdx.x * 16;

  // Wave-relative LDS byte offset of At, derived from its address so the array
  // escapes (alias analysis must assume the async asm writes it) and so the
  // offset is correct even if At is not at LDS offset 0.
  unsigned lds_base = (unsigned)(unsigned long long)(void*)At;

  // async copy 16x256 floats (1024 x 16B chunks) global -> padded LDS rows
  for (int f = threadIdx.x; f < 1024; f += 256) {
    int m = f >> 6, kc = f & 63;                 // row, 16B-chunk within row
    unsigned ldsb = lds_base + (unsigned)((m * 65 + kc) * 16);
    unsigned long long gp =
        (unsigned long long)(h + (size_t)rowbase * INDIM + (size_t)f * 4);
    asm volatile("global_load_async_to_lds_b128 %0, %1, off"
                 :: "v"(ldsb), "v"(gp) : "memory");
  }
  asm volatile("s_wait_asynccnt 0" ::: "memory");
  __syncthreads();

  int wave = threadIdx.x >> 5;
  int lane = threadIdx.x & 31;
  int mrow = lane & 15;            // A row (lanes 0-15), also B/C column index
  int khalf = (lane >> 4) * 2;     // 0 or 2 : K sub-block per ISA 16x16x4 layout
  int rofs = (lane >> 4) * 8;      // C rows: M = vgpr + (lane>=16 ? 8 : 0)

  for (int t = 0; t < 3; ++t) {
    int colbase = (wave + t * 8) * 16;     // 24 column tiles across 8 waves
    // K-pair interleaved B: lane's pair lives at Wp[((k+khalf)>>1)*768 + col*2]
    const float* bp = Wp + (size_t)(colbase + mrow) * 2 + (khalf >> 1) * (OC * 2);
    v8f acc = {};
#pragma unroll 8
    for (int k = 0; k < INDIM; k += 4) {
      v2f a = *(const v2f*)&At[mrow * 260 + k + khalf];          // ds_load_b64
      v2f b = *(const v2f*)&bp[(size_t)(k >> 1) * (OC * 2)];     // global b64
      acc = __builtin_amdgcn_wmma_f32_16x16x4_f32(false, a, false, b,
                                                  (short)0, acc, false, false);
    }
#pragma unroll
    for (int i = 0; i < 8; ++i)
      y[(size_t)(rowbase + rofs + i) * OC + colbase + mrow] = acc[i];
  }
}

// ---------------- edge logits: a = ELU(sum((q - c) * c)), seg-max via uint key ----
__global__ void edge_kernel(const float* __restrict__ y, const int* __restrict__ src,
                            const int* __restrict__ dst, float* __restrict__ a,
                            unsigned* __restrict__ mkey) {
  int gid = blockIdx.x * blockDim.x + threadIdx.x;
  if (gid >= EE * HH) return;
  int e = gid >> 2, hh = gid & 3;
  int s = src[e], d = dst[e];
  const float4* c4 = (const float4*)(y + (size_t)d * OC + 128 + hh * 32);
  const float4* q4 = (const float4*)(y + (size_t)s * OC + 256 + hh * 32);
  float sum = 0.0f;
#pragma unroll
  for (int i = 0; i < 8; ++i) {
    float4 c = c4[i], q = q4[i];
    sum += (q.x - c.x) * c.x + (q.y - c.y) * c.y +
           (q.z - c.z) * c.z + (q.w - c.w) * c.w;
  }
  float av = sum > 0.0f ? sum : expm1f(sum);   // ELU(alpha=1)
  a[gid] = av;
  atomicMax(&mkey[(size_t)d * HH + hh], fkey(av));   // order-independent -> deterministic
}

// ---------------- exp(a - m), seg-sum ----------------
__global__ void exp_kernel(const int* __restrict__ dst, float* __restrict__ a,
                           const unsigned* __restrict__ mkey, float* __restrict__ ssum) {
  int gid = blockIdx.x * blockDim.x + threadIdx.x;
  if (gid >= EE * HH) return;
  int e = gid >> 2, hh = gid & 3;
  int d = dst[e];
  float m = fkey_inv(mkey[(size_t)d * HH + hh]);
  float ex = expf(a[gid] - m);
  a[gid] = ex;
  atomicAdd(&ssum[(size_t)d * HH + hh], ex);
}

// ---------------- normalize -> attn (in place) ----------------
__global__ void norm_kernel(const int* __restrict__ dst, float* __restrict__ a,
                            const float* __restrict__ ssum) {
  int gid = blockIdx.x * blockDim.x + threadIdx.x;
  if (gid >= EE * HH) return;
  int e = gid >> 2, hh = gid & 3;
  a[gid] = a[gid] / ssum[(size_t)dst[e] * HH + hh];
}

// ---------------- radix select over raw float bits (all attn > 0) ----------------
__global__ void hist_kernel(const float* __restrict__ attn, Ctl* ctl, int pass) {
  __shared__ unsigned lh[1024];
  for (int i = threadIdx.x; i < 1024; i += blockDim.x) lh[i] = 0u;
  __syncthreads();
  unsigned pfx[4];
#pragma unroll
  for (int h = 0; h < 4; ++h) pfx[h] = ctl->prefix[h];
  int shift = 24 - 8 * pass;
  int total = EE * HH;
  for (int i = blockIdx.x * blockDim.x + threadIdx.x; i < total;
       i += gridDim.x * blockDim.x) {
    unsigned key = __float_as_uint(attn[i]);
    int h = i & 3;
    bool match = (pass == 0) || ((key >> (shift + 8)) == (pfx[h] >> (shift + 8)));
    if (match) atomicAdd(&lh[h * 256 + ((key >> shift) & 255u)], 1u);
  }
  __syncthreads();
  for (int i = threadIdx.x; i < 1024; i += blockDim.x)
    if (lh[i]) atomicAdd(&ctl->hist[i], lh[i]);
}

__global__ void scan_kernel(Ctl* ctl, int pass) {
  int shift = 24 - 8 * pass;
  if (threadIdx.x < 4) {
    int h = threadIdx.x;
    unsigned krem = ctl->krem[h];
    unsigned cum = 0, sel = 255;
    for (int b = 0; b < 256; ++b) {
      unsigned c = ctl->hist[h * 256 + b];
      if (cum + c >= krem) { sel = (unsigned)b; break; }
      cum += c;
    }
    ctl->prefix[h] |= sel << shift;
    ctl->krem[h] = krem - cum;
  }
  __syncthreads();
  for (int i = threadIdx.x; i < 1024; i += blockDim.x) ctl->hist[i] = 0u;  // next pass
}

// ---------------- sums below threshold + totals ----------------
__global__ void sumpass_kernel(const float* __restrict__ attn, Ctl* ctl) {
  __shared__ float sl[4], st[4];
  __shared__ unsigned cl[4];
  if (threadIdx.x < 4) { sl[threadIdx.x] = 0.f; st[threadIdx.x] = 0.f; cl[threadIdx.x] = 0u; }
  __syncthreads();
  int h = threadIdx.x & 3;
  float thd = __uint_as_float(ctl->prefix[h]);
  float a_tot = 0.f, a_lt = 0.f;
  unsigned c_lt = 0u;
  int q = (blockIdx.x * blockDim.x + threadIdx.x) >> 2;
  int qs = (gridDim.x * blockDim.x) >> 2;
  for (int e = q; e < EE; e += qs) {
    float v = attn[e * HH + h];
    a_tot += v;
    if (v < thd) { a_lt += v; ++c_lt; }
  }
  atomicAdd(&st[h], a_tot);
  atomicAdd(&sl[h], a_lt);
  atomicAdd(&cl[h], c_lt);
  __syncthreads();
  if (threadIdx.x < 4) {
    atomicAdd(&ctl->sum_tot[threadIdx.x], st[threadIdx.x]);
    atomicAdd(&ctl->sum_lt[threadIdx.x], sl[threadIdx.x]);
    atomicAdd(&ctl->cnt_lt[threadIdx.x], cl[threadIdx.x]);
  }
}

__global__ void finalize_kernel(Ctl* ctl) {
  int h = threadIdx.x;
  if (h < 4) {
    float thd = __uint_as_float(ctl->prefix[h]);
    float topk = ctl->sum_lt[h] + (float)(LSK - ctl->cnt_lt[h]) * thd;  // ties -> thd copies
    float s2 = ctl->sum_tot[h] - ctl->sum_lt[h];                        // sum(a2)
    ctl->ratio[h] = (s2 + topk) / s2;
  }
}

// ---------------- weighted scatter: out[dst, h*32+d] += ft[src] * a3 ----------------
__global__ void scatter_kernel(const float* __restrict__ y, const float* __restrict__ attn,
                               const int* __restrict__ src, const int* __restrict__ dst,
                               const Ctl* __restrict__ ctl, float* __restrict__ out) {
  int gid = blockIdx.x * blockDim.x + threadIdx.x;
  if (gid >= EE * 128) return;
  int e = gid >> 7, col = gid & 127, h = col >> 5;
  float v = attn[e * HH + h];
  float thd = __uint_as_float(ctl->prefix[h]);
  if (v >= thd) {   // attn - thd < 0 ? 0 : attn
    float w = v * ctl->ratio[h];
    atomicAdd(&out[(size_t)dst[e] * OUTC + col],
              y[(size_t)src[e] * OC + col] * w);
  }
}

// ---------------- mean over heads of ft -> cols 128..159 ----------------
__global__ void mean_kernel(const float* __restrict__ y, float* __restrict__ out) {
  int gid = blockIdx.x * blockDim.x + threadIdx.x;
  if (gid >= NN * 32) return;
  int n = gid >> 5, d2 = gid & 31;
  const float* r = y + (size_t)n * OC;
  out[(size_t)n * OUTC + 128 + d2] =
      0.25f * (r[d2] + r[32 + d2] + r[64 + d2] + r[96 + d2]);
}

extern "C" void kernel_launch(void* const* d_in, const int* in_sizes, int n_in,
                              void* d_out, int out_size, void* d_ws, size_t ws_size,
                              hipStream_t stream) {
  const float* h  = (const float*)d_in[0];
  const float* Wl = (const float*)d_in[1];
  const float* fc = (const float*)d_in[2];
  const float* fq = (const float*)d_in[3];
  const int* src  = (const int*)d_in[4];
  const int* dst  = (const int*)d_in[5];
  float* out = (float*)d_out;

  char* ws = (char*)d_ws;
  size_t o_wp   = 0;                               // 256*384*4   = 393,216
  size_t o_y    = o_wp + 393216;                   // 50000*384*4 = 76,800,000
  size_t o_a    = o_y + 76800000;                  // 800000*4*4  = 12,800,000
  size_t o_mkey = o_a + 12800000;                  // 50000*4*4   = 800,000
  size_t o_ssum = o_mkey + 800000;                 // 800,000
  size_t o_ctl  = o_ssum + 800000;
  float*    Wp   = (float*)(ws + o_wp);
  float*    y    = (float*)(ws + o_y);
  float*    a    = (float*)(ws + o_a);
  unsigned* mkey = (unsigned*)(ws + o_mkey);
  float*    ssum = (float*)(ws + o_ssum);
  Ctl*      ctl  = (Ctl*)(ws + o_ctl);

  init_kernel<<<8192, 256, 0, stream>>>(out, mkey, ssum, ctl);
  pack_kernel<<<(INDIM * OC + 255) / 256, 256, 0, stream>>>(Wl, fc, fq, Wp);
  gemm_kernel<<<NN / 16, 256, 0, stream>>>(h, Wp, y);

  int egrid = (EE * HH + 255) / 256;
  edge_kernel<<<egrid, 256, 0, stream>>>(y, src, dst, a, mkey);
  exp_kernel<<<egrid, 256, 0, stream>>>(dst, a, mkey, ssum);
  norm_kernel<<<egrid, 256, 0, stream>>>(dst, a, ssum);

  for (int pass = 0; pass < 4; ++pass) {
    hist_kernel<<<2048, 256, 0, stream>>>(a, ctl, pass);
    scan_kernel<<<1, 256, 0, stream>>>(ctl, pass);
  }
  sumpass_kernel<<<2048, 256, 0, stream>>>(a, ctl);
  finalize_kernel<<<1, 32, 0, stream>>>(ctl);

  scatter_kernel<<<(EE * 128 + 255) / 256, 256, 0, stream>>>(y, a, src, dst, ctl, out);
  mean_kernel<<<(NN * 32 + 255) / 256, 256, 0, stream>>>(y, out);
}